// GCNModel_61538291417125
// MI455X (gfx1250) — compile-verified
//
#include <hip/hip_runtime.h>
#include <hip/hip_bf16.h>

typedef __attribute__((ext_vector_type(2))) float v2f;
typedef __attribute__((ext_vector_type(8))) float v8f;

#define D_FEAT 128   // D_IN = D_H1 = D_H2
#define N_CLS   64

// ---------------------------------------------------------------------------
// Degree / normalization kernels
// ---------------------------------------------------------------------------
__global__ void init_deg_kernel(float* deg, int n) {
    int i = blockIdx.x * blockDim.x + threadIdx.x;
    if (i < n) deg[i] = 1.0f;               // self-loop contributes 1
}

__global__ void edge_deg_kernel(const long long* __restrict__ ei, float* deg,
                                int E) {
    int e = blockIdx.x * blockDim.x + threadIdx.x;
    if (e < E) {
        long long t = ei[(size_t)E + e];    // dst row of edge_index
        atomicAdd(&deg[t], 1.0f);
    }
}

__global__ void rsqrt_inplace_kernel(float* deg, int n) {
    int i = blockIdx.x * blockDim.x + threadIdx.x;
    if (i < n) {
        float d = deg[i];
        deg[i] = (d > 0.0f) ? rsqrtf(d) : 0.0f;   // d >= 1 always, but match ref
    }
}

// ---------------------------------------------------------------------------
// WMMA f32 GEMM:  C[M,NCOLS] = A[M,KDIM] @ B[KDIM,NCOLS] (+ bias if HAS_BIAS)
// Compile-time NCOLS/KDIM -> fully unrolled K-loop, all loads are
// immediate-offset from per-lane base pointers (IOFFSET is 24-bit signed;
// max offset here is 125*128*4 = 64000 B).  One wave per 16x16 tile,
// 32x V_WMMA_F32_16X16X4_F32 per tile.  Requires M % 16 == 0.
// ---------------------------------------------------------------------------
template <int NCOLS, int KDIM, bool HAS_BIAS>
__global__ void gemm_wmma_f32(const float* __restrict__ A,
                              const float* __restrict__ B,
                              const float* __restrict__ bias,
                              float* __restrict__ C,
                              int M) {
    const int wavesPerBlock = blockDim.x >> 5;
    const int wv   = threadIdx.x >> 5;
    const int lane = threadIdx.x & 31;
    const int tile = blockIdx.x * wavesPerBlock + wv;

    constexpr int NT = NCOLS >> 4;
    const int total  = (M >> 4) * NT;
    if (tile >= total) return;              // uniform per wave -> EXEC all 1s

    const int tm   = tile / NT;
    const int tn   = tile % NT;
    const int half = lane >> 4;             // 0: lanes 0-15, 1: lanes 16-31
    const int r    = lane & 15;
    const int bcol = tn * 16 + r;

    // Per-lane bases; all K-loop accesses become base[compile-time-imm].
    const float* __restrict__ Abase =
        A + (size_t)(tm * 16 + r) * KDIM + 2 * half;
    const float* __restrict__ Bbase =
        B + (size_t)(2 * half) * NCOLS + bcol;

    v8f acc = {};
#pragma unroll
    for (int k0 = 0; k0 < KDIM; k0 += 4) {
        v2f a, b;
        // A 16x4 fragment: lane holds A[row][k0+2h], A[row][k0+2h+1]
        a.x = Abase[k0];
        a.y = Abase[k0 + 1];
        // B 4x16 fragment: lane holds B[k0+2h][col], B[k0+2h+1][col]
        b.x = Bbase[(size_t)k0 * NCOLS];
        b.y = Bbase[(size_t)(k0 + 1) * NCOLS];
        acc = __builtin_amdgcn_wmma_f32_16x16x4_f32(
            /*neg_a=*/false, a, /*neg_b=*/false, b,
            /*c_mod=*/(short)0, acc, /*reuse_a=*/false, /*reuse_b=*/false);
    }

    float bv = 0.0f;
    if constexpr (HAS_BIAS) bv = bias[bcol];

    float* __restrict__ Cbase =
        C + (size_t)(tm * 16 + half * 8) * NCOLS + bcol;
#pragma unroll
    for (int i = 0; i < 8; ++i) {           // C/D layout: M = i + 8*half
        Cbase[(size_t)i * NCOLS] = acc[i] + bv;
    }
}

// ---------------------------------------------------------------------------
// Edge scatter:  agg[dst] += h[src] * dinv[src]*dinv[dst]
// One wave per edge; each lane handles 4 contiguous floats (128 = 32*4),
// so loads are a coalesced 512B burst and atomics hit full 128B sectors.
// Everything is L2-resident (h = 25.6 MB << 192 MB L2).
// ---------------------------------------------------------------------------
__global__ void edge_scatter_kernel(const long long* __restrict__ ei,
                                    const float* __restrict__ dinv,
                                    const float* __restrict__ h,
                                    float* __restrict__ agg,
                                    int E) {
    const int lane = threadIdx.x & 31;
    const int wv   = threadIdx.x >> 5;
    const int e    = blockIdx.x * (blockDim.x >> 5) + wv;
    if (e >= E) return;

    const long long s = ei[e];
    const long long t = ei[(size_t)E + e];
    const float w = dinv[s] * dinv[t];

    const float4* __restrict__ hv =
        reinterpret_cast<const float4*>(h + (size_t)s * D_FEAT);
    float4 v = hv[lane];

    float* ag = agg + (size_t)t * D_FEAT + lane * 4;
    atomicAdd(ag + 0, v.x * w);
    atomicAdd(ag + 1, v.y * w);
    atomicAdd(ag + 2, v.z * w);
    atomicAdd(ag + 3, v.w * w);
}

// ---------------------------------------------------------------------------
// Epilogue: add self-loop message, bias, ReLU (in place on agg)
//   agg[i][d] = relu(agg[i][d] + h[i][d]*dinv[i]^2 + bias[d])
// ---------------------------------------------------------------------------
__global__ void selfloop_bias_relu_kernel(float* __restrict__ agg,
                                          const float* __restrict__ h,
                                          const float* __restrict__ dinv,
                                          const float* __restrict__ bias,
                                          int nNodes) {
    const size_t idx = (size_t)blockIdx.x * blockDim.x + threadIdx.x;
    const size_t tot = (size_t)nNodes * D_FEAT;
    if (idx >= tot) return;
    const int i = (int)(idx >> 7);          // / 128
    const int d = (int)(idx & (D_FEAT - 1));
    const float di = dinv[i];
    float v = agg[idx] + h[idx] * di * di + bias[d];
    agg[idx] = v > 0.0f ? v : 0.0f;
}

// ---------------------------------------------------------------------------
// Launcher
// ---------------------------------------------------------------------------
extern "C" void kernel_launch(void* const* d_in, const int* in_sizes, int n_in,
                              void* d_out, int out_size, void* d_ws, size_t ws_size,
                              hipStream_t stream) {
    const float*     x  = (const float*)d_in[0];
    const long long* ei = (const long long*)d_in[1];   // int64 [2][E]
    const float*     W1 = (const float*)d_in[2];
    const float*     b1 = (const float*)d_in[3];
    const float*     W2 = (const float*)d_in[4];
    const float*     b2 = (const float*)d_in[5];
    const float*     Wl = (const float*)d_in[6];
    const float*     bl = (const float*)d_in[7];
    float* out = (float*)d_out;

    const int N = in_sizes[0] / D_FEAT;      // 50000 (multiple of 16)
    const int E = in_sizes[1] / 2;           // 640000

    // Workspace layout: dinv | bufA | bufB
    char* ws = (char*)d_ws;
    size_t off0 = ((size_t)N * sizeof(float) + 255) & ~(size_t)255;
    float* dinv = (float*)ws;
    float* bufA = (float*)(ws + off0);                        // 25.6 MB
    float* bufB = bufA + (size_t)N * D_FEAT;                  // 25.6 MB
    const size_t featBytes = (size_t)N * D_FEAT * sizeof(float);

    // ---- degree -> dinv --------------------------------------------------
    init_deg_kernel<<<(N + 255) / 256, 256, 0, stream>>>(dinv, N);
    edge_deg_kernel<<<(E + 255) / 256, 256, 0, stream>>>(ei, dinv, E);
    rsqrt_inplace_kernel<<<(N + 255) / 256, 256, 0, stream>>>(dinv, N);

    const int wavesPerBlock = 8;             // 256 threads = 8 wave32
    auto gemmBlocks = [&](int M, int Ncols) {
        int tiles = (M >> 4) * (Ncols >> 4);
        return (tiles + wavesPerBlock - 1) / wavesPerBlock;
    };
    const int edgeBlocks = (E + wavesPerBlock - 1) / wavesPerBlock;
    const int elemBlocks = (int)(((size_t)N * D_FEAT + 255) / 256);

    // ---- layer 1: h1 = x@W1 ; agg1 = scatter(h1) ; relu(+self,+b1) -------
    gemm_wmma_f32<D_FEAT, D_FEAT, false><<<gemmBlocks(N, D_FEAT), 256, 0, stream>>>(
        x, W1, nullptr, bufA, N);
    hipMemsetAsync(bufB, 0, featBytes, stream);
    edge_scatter_kernel<<<edgeBlocks, 256, 0, stream>>>(ei, dinv, bufA, bufB, E);
    selfloop_bias_relu_kernel<<<elemBlocks, 256, 0, stream>>>(bufB, bufA, dinv, b1, N);

    // ---- layer 2: h2 = h1r@W2 ; agg2 = scatter(h2) ; relu(+self,+b2) -----
    gemm_wmma_f32<D_FEAT, D_FEAT, false><<<gemmBlocks(N, D_FEAT), 256, 0, stream>>>(
        bufB, W2, nullptr, bufA, N);
    hipMemsetAsync(bufB, 0, featBytes, stream);
    edge_scatter_kernel<<<edgeBlocks, 256, 0, stream>>>(ei, dinv, bufA, bufB, E);
    selfloop_bias_relu_kernel<<<elemBlocks, 256, 0, stream>>>(bufB, bufA, dinv, b2, N);

    // ---- head: out = h2r @ Wl + bl --------------------------------------
    gemm_wmma_f32<N_CLS, D_FEAT, true><<<gemmBlocks(N, N_CLS), 256, 0, stream>>>(
        bufB, Wl, bl, out, N);
}